// FAPELoss_84138409328675
// MI455X (gfx1250) — compile-verified
//
#include <hip/hip_runtime.h>
#include <math.h>

// ---------------------------------------------------------------------------
// FAPE loss on gfx1250 (MI455X), wave32 + V_WMMA_F32_16X16X4_F32.
//
// Pair transform as homogeneous rank-4 GEMM, two WMMAs per component:
//   D_k[a, f] = [p_a,1] . Bp_k[f]  +  [q_a,1] . Bq_k[f]
// where Bp_k column f = ( R_p[:,k], -(t_p . R_p[:,k]) )   (4 rows)
//       Bq_k column f = (-R_t[:,k], +(t_t . R_t[:,k]) )
// 6 WMMAs -> 16 atoms x 16 frames of full diff vectors; per-lane
// sqrt(d0^2+d1^2+d2^2), clamp, mask, accumulate. No cross-lane traffic.
// ---------------------------------------------------------------------------

typedef __attribute__((ext_vector_type(2))) float v2f;
typedef __attribute__((ext_vector_type(4))) float v4f;
typedef __attribute__((ext_vector_type(8))) float v8f;

#define FAPE_EPS    1e-8f
#define FAPE_CLAMP  10.0f

// -------------------------------------------------------------------------
// Phase 1: per-frame B-column records, 24 floats, grouped by B-row half so
// each lane's 12 floats are contiguous (3x b128 loads in phase 2):
//   rec[h*12 + 0.. 5] : pred side, [k0r0,k0r1, k1r0,k1r1, k2r0,k2r1]
//   rec[h*12 + 6..11] : true side, same order (sign folded)
// where h=0 covers B rows {0,1}, h=1 covers B rows {2,3}; row 3 = bias.
// -------------------------------------------------------------------------
__device__ inline void frame_record(const float* __restrict__ a /*N,CA,C xyz*/,
                                    float* __restrict__ rec /*24 floats*/,
                                    int side /*0=pred,1=true*/, float sgn) {
    const float nx = a[0], ny = a[1], nz = a[2];   // N
    const float tx = a[3], ty = a[4], tz = a[5];   // CA (= translation)
    const float cx = a[6], cy = a[7], cz = a[8];   // C

    float v1x = cx - tx, v1y = cy - ty, v1z = cz - tz;
    float v2x = nx - tx, v2y = ny - ty, v2z = nz - tz;

    float inv1 = 1.0f / sqrtf(v1x*v1x + v1y*v1y + v1z*v1z + FAPE_EPS);
    float e1x = v1x*inv1, e1y = v1y*inv1, e1z = v1z*inv1;

    float d = v2x*e1x + v2y*e1y + v2z*e1z;
    float ux = v2x - d*e1x, uy = v2y - d*e1y, uz = v2z - d*e1z;
    float inv2 = 1.0f / sqrtf(ux*ux + uy*uy + uz*uz + FAPE_EPS);
    float e2x = ux*inv2, e2y = uy*inv2, e2z = uz*inv2;

    float e3x = e1y*e2z - e1z*e2y;
    float e3y = e1z*e2x - e1x*e2z;
    float e3z = e1x*e2y - e1y*e2x;

    const float ex[3] = {e1x, e2x, e3x};
    const float ey[3] = {e1y, e2y, e3y};
    const float ez[3] = {e1z, e2z, e3z};
    #pragma unroll
    for (int k = 0; k < 3; ++k) {
        // B column rows 0..3 for component k (sign folded for true side)
        float col0 = sgn * ex[k];
        float col1 = sgn * ey[k];
        float col2 = sgn * ez[k];
        float col3 = -sgn * (tx*ex[k] + ty*ey[k] + tz*ez[k]);
        rec[0*12 + side*6 + k*2 + 0] = col0;   // half 0: rows 0,1
        rec[0*12 + side*6 + k*2 + 1] = col1;
        rec[1*12 + side*6 + k*2 + 0] = col2;   // half 1: rows 2,3
        rec[1*12 + side*6 + k*2 + 1] = col3;
    }
}

__global__ __launch_bounds__(256) void fape_build_frames(
    const float* __restrict__ pred, const float* __restrict__ truec,
    float* __restrict__ rec, int total) {
    int i = blockIdx.x * blockDim.x + threadIdx.x;   // i = b*Nres + r
    if (i >= total) return;
    frame_record(pred  + (size_t)i * 9, rec + (size_t)i * 24, 0,  1.0f);
    frame_record(truec + (size_t)i * 9, rec + (size_t)i * 24, 1, -1.0f);
}

// -------------------------------------------------------------------------
// Phase 2: all-pairs clamped FAPE error. One block per (batch, 16-atom
// tile); 8 waves stride over 16-frame groups. 6 WMMAs -> 256 pairs.
// -------------------------------------------------------------------------
__global__ __launch_bounds__(256) void fape_pair_kernel(
    const float* __restrict__ predCoords,
    const float* __restrict__ trueCoords,
    const float* __restrict__ atomMask,
    const float* __restrict__ frameRec,
    float* __restrict__ blockPartials,
    int Nres) {
    const int tid  = threadIdx.x;
    const int lane = tid & 31;
    const int wave = tid >> 5;
    const int tilesPerBatch = Nres >> 4;
    const int b        = blockIdx.x / tilesPerBatch;
    const int atomBase = (blockIdx.x % tilesPerBatch) << 4;

    const int half = lane >> 4;     // A: K rows {0,1} vs {2,3}; B: rows {0,1} vs {2,3}
    const int sub  = lane & 15;     // A: row M;  B/D: column N

    // ---- A matrices (fixed per wave): rows = 16 atoms, cols = [x,y,z,1] ----
    const int aAtom = atomBase + sub;
    const float* pca = predCoords + ((size_t)(b * Nres + aAtom) * 3 + 1) * 3;
    const float* qca = trueCoords + ((size_t)(b * Nres + aAtom) * 3 + 1) * 3;
    v2f Ap, Aq;
    if (half == 0) { Ap[0] = pca[0]; Ap[1] = pca[1]; Aq[0] = qca[0]; Aq[1] = qca[1]; }
    else           { Ap[0] = pca[2]; Ap[1] = 1.0f;   Aq[0] = qca[2]; Aq[1] = 1.0f;  }

    // ---- per-row atom masks for the C/D tile this lane owns ----
    float ma[8];
    #pragma unroll
    for (int v = 0; v < 8; ++v)
        ma[v] = atomMask[(size_t)(b * Nres + atomBase + v + half * 8) * 3 + 1];

    const float* recBase = frameRec + (size_t)b * Nres * 24 + half * 12;
    const int nGroups = Nres >> 4;          // Nres % 16 == 0 in this workload
    float acc = 0.0f;

    for (int g = wave; g < nGroups; g += 8) {
        const int f = g * 16 + sub;         // this lane's frame (column N)
        const float* rec = recBase + (size_t)f * 24;

        // 12 contiguous floats: Bp rows for k=0..2, then Bq rows for k=0..2
        v4f c0 = *(const v4f*)(rec + 0);
        v4f c1 = *(const v4f*)(rec + 4);
        v4f c2 = *(const v4f*)(rec + 8);
        v2f Bp0 = {c0[0], c0[1]}, Bp1 = {c0[2], c0[3]}, Bp2 = {c1[0], c1[1]};
        v2f Bq0 = {c1[2], c1[3]}, Bq1 = {c2[0], c2[1]}, Bq2 = {c2[2], c2[3]};

        float mf = atomMask[(size_t)(b * Nres + f) * 3 + 1];

        v8f d0 = {}, d1 = {}, d2 = {};
        d0 = __builtin_amdgcn_wmma_f32_16x16x4_f32(false, Ap, false, Bp0,
                                                   (short)0, d0, false, false);
        d1 = __builtin_amdgcn_wmma_f32_16x16x4_f32(false, Ap, false, Bp1,
                                                   (short)0, d1, false, false);
        d2 = __builtin_amdgcn_wmma_f32_16x16x4_f32(false, Ap, false, Bp2,
                                                   (short)0, d2, false, false);
        d0 = __builtin_amdgcn_wmma_f32_16x16x4_f32(false, Aq, false, Bq0,
                                                   (short)0, d0, false, false);
        d1 = __builtin_amdgcn_wmma_f32_16x16x4_f32(false, Aq, false, Bq1,
                                                   (short)0, d1, false, false);
        d2 = __builtin_amdgcn_wmma_f32_16x16x4_f32(false, Aq, false, Bq2,
                                                   (short)0, d2, false, false);

        // per-lane full diff vectors: 8 atoms x this lane's frame.
        // raw v_sqrt_f32 (~1 ulp) -- skips libm's range-scale + Newton fixup.
        float inner = 0.0f;
        #pragma unroll
        for (int v = 0; v < 8; ++v) {
            float s = d0[v]*d0[v] + d1[v]*d1[v] + d2[v]*d2[v];
            float e = fminf(__builtin_amdgcn_sqrtf(s + FAPE_EPS), FAPE_CLAMP);
            inner += e * ma[v];
        }
        acc += mf * inner;
    }

    // deterministic block tree-reduction
    __shared__ float red[256];
    red[tid] = acc;
    __syncthreads();
    #pragma unroll
    for (int s = 128; s > 0; s >>= 1) {
        if (tid < s) red[tid] += red[tid + s];
        __syncthreads();
    }
    if (tid == 0) blockPartials[blockIdx.x] = red[0];
}

// -------------------------------------------------------------------------
// Phase 3: deterministic serial finalize (tiny).
// -------------------------------------------------------------------------
__global__ void fape_finalize(const float* __restrict__ partials,
                              const float* __restrict__ atomMask,
                              float* __restrict__ out,
                              int Nres, int tilesPerBatch, int B) {
    if (blockIdx.x != 0 || threadIdx.x != 0) return;
    float total = 0.0f;
    for (int b = 0; b < B; ++b) {
        float num = 0.0f;
        for (int t = 0; t < tilesPerBatch; ++t)
            num += partials[b * tilesPerBatch + t];
        float S = 0.0f;
        for (int r = 0; r < Nres; ++r)
            S += atomMask[(size_t)(b * Nres + r) * 3 + 1];
        total += num / (S * S + FAPE_EPS);   // sum(pair_mask) == (sum m)^2
    }
    *out = total / (float)B;
}

// -------------------------------------------------------------------------
extern "C" void kernel_launch(void* const* d_in, const int* in_sizes, int n_in,
                              void* d_out, int out_size, void* d_ws, size_t ws_size,
                              hipStream_t stream) {
    const float* pred  = (const float*)d_in[0];
    const float* truec = (const float*)d_in[1];
    const float* mask  = (const float*)d_in[2];
    float* out = (float*)d_out;

    const int B    = 4;
    const int Nres = in_sizes[0] / (B * 9);     // B*Nres*3atoms*3xyz
    const int tilesPerBatch = Nres / 16;
    const int nPairBlocks   = B * tilesPerBatch;
    const int totalFrames   = B * Nres;

    float* wsFrames   = (float*)d_ws;                        // B*Nres*24 floats
    float* wsPartials = wsFrames + (size_t)totalFrames * 24; // nPairBlocks floats

    hipLaunchKernelGGL(fape_build_frames,
                       dim3((totalFrames + 255) / 256), dim3(256), 0, stream,
                       pred, truec, wsFrames, totalFrames);

    hipLaunchKernelGGL(fape_pair_kernel,
                       dim3(nPairBlocks), dim3(256), 0, stream,
                       pred, truec, mask, wsFrames, wsPartials, Nres);

    hipLaunchKernelGGL(fape_finalize,
                       dim3(1), dim3(32), 0, stream,
                       wsPartials, mask, out, Nres, tilesPerBatch, B);
}